// TorchWelfordEstimator_32555852103908
// MI455X (gfx1250) — compile-verified
//
#include <hip/hip_runtime.h>
#include <stdint.h>

// Problem geometry (fixed by the reference): x is [B, C, H, W] f32.
#define B_DIM          32
#define CHW            (64 * 128 * 128)   // 1,048,576 independent positions
#define POS_PER_BLOCK  1024               // positions owned by one workgroup
#define TPB            256                // 8 wave32s
#define HALF_B         16                 // batch double-buffer split
#define EPT            4                  // float4 per thread

typedef uint32_t v4u __attribute__((ext_vector_type(4)));
typedef int      v8i __attribute__((ext_vector_type(8)));
typedef int      v4i __attribute__((ext_vector_type(4)));

// ---------------------------------------------------------------------------
// Tensor Data Mover: async DMA of a 2D tile (tile_d0 contiguous elems x
// tile_d1 rows, row stride d0_stride elems) from global -> LDS.
// Descriptor layout per CDNA5 ISA (D# groups 0/1).
// ---------------------------------------------------------------------------
__device__ __forceinline__ void tdm_load_tile_2d(uint32_t lds_byte_off,
                                                 const void* global_tile_start,
                                                 uint32_t tile_d0, uint32_t tile_d1,
                                                 uint32_t tensor_d0, uint32_t tensor_d1,
                                                 uint64_t d0_stride_elems)
{
    uint64_t ga = (uint64_t)(uintptr_t)global_tile_start;

    v4u g0;
    g0.x = 1u;                                    // count=1 (valid user descriptor)
    g0.y = lds_byte_off;                          // LDS destination address (bytes)
    g0.z = (uint32_t)ga;                          // global_addr[31:0]
    g0.w = (uint32_t)((ga >> 32) & 0x01FFFFFFu)   // global_addr[56:32]
         | (2u << 30);                            // type = 2 ("image")

    v8i g1;
    g1[0] = (int)(2u << 16);                                  // data_size = 2 -> 4 bytes
    g1[1] = (int)((tensor_d0 & 0xFFFFu) << 16);               // tensor_dim0[15:0]
    g1[2] = (int)(((tensor_d0 >> 16) & 0xFFFFu)               // tensor_dim0[31:16]
                 | ((tensor_d1 & 0xFFFFu) << 16));            // tensor_dim1[15:0]
    g1[3] = (int)(((tensor_d1 >> 16) & 0xFFFFu)               // tensor_dim1[31:16]
                 | ((tile_d0 & 0xFFFFu) << 16));              // tile_dim0
    g1[4] = (int)(tile_d1 & 0xFFFFu);                         // tile_dim1 (tile_dim2=0)
    g1[5] = (int)(d0_stride_elems & 0xFFFFFFFFu);             // tensor_dim0_stride lo
    g1[6] = (int)((d0_stride_elems >> 32) & 0xFFFFu);         // stride hi (dim1_stride=0)
    g1[7] = 0;

    v4i gz = {0, 0, 0, 0};
#if __clang_major__ >= 23
    v8i gz8 = {0, 0, 0, 0, 0, 0, 0, 0};
    __builtin_amdgcn_tensor_load_to_lds(g0, g1, gz, gz, gz8, 0);
#else
    __builtin_amdgcn_tensor_load_to_lds(g0, g1, gz, gz, 0);
#endif
}

// One Welford step on a single lane value (matches the reference recurrence;
// the uniform divide 1/(n+1) is hoisted per batch step by the caller).
__device__ __forceinline__ void welford_step(float x, float& m, float& s, int& nz,
                                             float inv_nf)
{
    nz += (x != 0.0f) ? 1 : 0;
    float d_old = x - m;              // xi - m_old
    float m_new = m + d_old * inv_nf; // m + (xi - m)/(n+1)
    s = fmaf(x - m_new, d_old, s);    // s += (xi - m_new)*(xi - m_old)
    m = m_new;
}

__global__ void __launch_bounds__(TPB)
welford_tdm_kernel(const float* __restrict__ x,
                   const float* __restrict__ m0,
                   const float* __restrict__ s0,
                   const int*   __restrict__ nz0,
                   const int*   __restrict__ n0p,
                   float*       __restrict__ out)
{
    __shared__ float tile[B_DIM * POS_PER_BLOCK];   // 128 KiB (2 WGs/WGP fit in 320 KiB)

    const uint32_t t     = threadIdx.x;
    const uint32_t chunk = blockIdx.x * POS_PER_BLOCK;
    const uint32_t g     = chunk + t * EPT;

    // Wave 0 only issues the DMAs. TDM ignores EXEC, so gate with a
    // wave-uniform (readfirstlane) condition -> scalar branch skips waves 1-7.
    const int waveId = __builtin_amdgcn_readfirstlane((int)(t >> 5));
    if (waveId == 0) {
        uint32_t lds_base = (uint32_t)(uintptr_t)(void*)&tile[0];
        // Batches 0..15 into first 64 KiB, 16..31 into second 64 KiB.
        tdm_load_tile_2d(lds_base,
                         x + chunk,
                         POS_PER_BLOCK, HALF_B, CHW, B_DIM, CHW);
        tdm_load_tile_2d(lds_base + HALF_B * POS_PER_BLOCK * 4u,
                         x + (size_t)HALF_B * CHW + chunk,
                         POS_PER_BLOCK, HALF_B, CHW, B_DIM, CHW);
    }

    // Independent state loads overlap the in-flight TDM traffic.
    float4 mv  = *(const float4*)(m0  + g);
    float4 sv  = *(const float4*)(s0  + g);
    int4   nzv = *(const int4*)  (nz0 + g);
    const int n0 = *n0p;

    // Half 0 resident (wave 0 has TENSORcnt=2; others pass with cnt 0).
    __builtin_amdgcn_s_wait_tensorcnt((short)1);
    __syncthreads();

    #pragma unroll
    for (int b = 0; b < HALF_B; ++b) {
        const float  inv = 1.0f / (float)(n0 + b + 1);   // wave-uniform
        const float4 xv  = *(const float4*)&tile[b * POS_PER_BLOCK + t * EPT];
        welford_step(xv.x, mv.x, sv.x, nzv.x, inv);
        welford_step(xv.y, mv.y, sv.y, nzv.y, inv);
        welford_step(xv.z, mv.z, sv.z, nzv.z, inv);
        welford_step(xv.w, mv.w, sv.w, nzv.w, inv);
    }

    // Half 1 resident.
    __builtin_amdgcn_s_wait_tensorcnt((short)0);
    __syncthreads();

    #pragma unroll
    for (int b = HALF_B; b < B_DIM; ++b) {
        const float  inv = 1.0f / (float)(n0 + b + 1);
        const float4 xv  = *(const float4*)&tile[b * POS_PER_BLOCK + t * EPT];
        welford_step(xv.x, mv.x, sv.x, nzv.x, inv);
        welford_step(xv.y, mv.y, sv.y, nzv.y, inv);
        welford_step(xv.z, mv.z, sv.z, nzv.z, inv);
        welford_step(xv.w, mv.w, sv.w, nzv.w, inv);
    }

    // d_out layout: [m (CHW) | s (CHW)] f32, then nz (CHW) i32, then n (1) i32.
    *(float4*)(out + g)       = mv;
    *(float4*)(out + CHW + g) = sv;
    int* outi = (int*)out;
    *(int4*)(outi + 2 * CHW + g) = nzv;
    if (blockIdx.x == 0 && t == 0) {
        outi[3 * CHW] = n0 + B_DIM;
    }
}

extern "C" void kernel_launch(void* const* d_in, const int* in_sizes, int n_in,
                              void* d_out, int out_size, void* d_ws, size_t ws_size,
                              hipStream_t stream)
{
    (void)in_sizes; (void)n_in; (void)out_size; (void)d_ws; (void)ws_size;
    const float* x   = (const float*)d_in[0];
    const float* m0  = (const float*)d_in[1];
    const float* s0  = (const float*)d_in[2];
    const int*   nz0 = (const int*)d_in[3];
    const int*   n0  = (const int*)d_in[4];
    float*       out = (float*)d_out;

    dim3 grid(CHW / POS_PER_BLOCK);   // 1024 workgroups
    welford_tdm_kernel<<<grid, TPB, 0, stream>>>(x, m0, s0, nz0, n0, out);
}